// kp_loss_57982058496385
// MI455X (gfx1250) — compile-verified
//
#include <hip/hip_runtime.h>
#include <hip/hip_bf16.h>
#include <math.h>

typedef float v2f __attribute__((ext_vector_type(2)));
typedef float v4f __attribute__((ext_vector_type(4)));
typedef float v8f __attribute__((ext_vector_type(8)));

#define BB 4
#define KK 16
#define HH 1024
#define WW 1024
#define BK (BB * KK)                     // 64 keypoints
#define COEF 0.15915494309189535f        // AMP / (2*pi*SIG^2)
#define INV2S2 0.5f                      // 1 / (2*SIG^2)

#define TPB 256
#define NBLK 8192
#define CPR (WW / 4)                              // 256 float4 chunks per row
#define TOTAL_CHUNKS (BK * HH * CPR)              // 2^24 chunks (2^26 floats)
#define STRIDE (NBLK * TPB)                       // 2^21 threads
#define ITERS (TOTAL_CHUNKS / STRIDE)             // 8 chunks per thread (exact)

// ds_swizzle group-of-32 SWAPX16: offset[15]=0, xor=0x10, or=0, and=0x1f
#define SWIZ_SWAPX16 0x401F

// ---------------------------------------------------------------------------
// Kernel 1: per-keypoint separable Gaussian factors (heatmap is rank-1).
//   FX[bk][i] = coef * exp(-((i+0.5) - cx)^2 * inv2s2)   (row factor, coef folded)
//   FY[bk][j] =        exp(-((j+0.5) - cy)^2 * inv2s2)   (column factor)
// with cx = target_x * W, cy = H * (1 - target_y).
// ---------------------------------------------------------------------------
__global__ __launch_bounds__(TPB) void kp_factors_kernel(
    const float* __restrict__ target, float* __restrict__ FX,
    float* __restrict__ FY) {
  const int bk = blockIdx.x;
  const float cx = target[bk * 2 + 0] * (float)WW;
  const float cy = (float)HH * (1.0f - target[bk * 2 + 1]);
  for (int i = threadIdx.x; i < HH; i += TPB) {
    const float g = (float)i + 0.5f;
    const float dx = g - cx;
    const float dy = g - cy;
    FX[bk * HH + i] = COEF * expf(-dx * dx * INV2S2);
    FY[bk * WW + i] = expf(-dy * dy * INV2S2);
  }
}

// ---------------------------------------------------------------------------
// Kernel 2: streaming squared-error reduction (memory-bound, ~11us floor at
// 23.3 TB/s for the 256 MB output read). Fully unrolled 8-chunk loop with 4
// independent accumulators; uniform control flow keeps EXEC all-1s.
// Wave32 reduction via V_WMMA_F32_16X16X4_F32 with B = ones:
//   per-lane A = {acc, 0}: lanes 0-15 -> A[m,0], lanes 16-31 -> A[m,2]
//   D[m,n] = acc_m + acc_{m+16}; per-lane sum of the 8 C/D VGPRs gives the
//   half-wave sums; one ds_swizzle SWAPX16 combines the two halves.
// ---------------------------------------------------------------------------
__global__ __launch_bounds__(TPB) void kp_reduce_kernel(
    const float* __restrict__ out, const float* __restrict__ FX,
    const float* __restrict__ FY, float* __restrict__ partial) {
  const unsigned tid = blockIdx.x * TPB + threadIdx.x;

  float a0 = 0.0f, a1 = 0.0f, a2 = 0.0f, a3 = 0.0f;
#pragma unroll
  for (int k = 0; k < ITERS; ++k) {
    const unsigned c = tid + (unsigned)k * STRIDE;
    const unsigned j4 = c & (CPR - 1);         // float4 index within row
    const unsigned i = (c >> 8) & (HH - 1);    // row
    const unsigned bk = c >> 18;               // keypoint
    const v4f o = __builtin_nontemporal_load((const v4f*)out + c);
    const float fx = FX[bk * HH + i];
    const v4f fy = *((const v4f*)(FY + bk * WW) + j4);
    const float d0 = o.x - fx * fy.x;
    const float d1 = o.y - fx * fy.y;
    const float d2 = o.z - fx * fy.z;
    const float d3 = o.w - fx * fy.w;
    a0 = fmaf(d0, d0, a0);
    a1 = fmaf(d1, d1, a1);
    a2 = fmaf(d2, d2, a2);
    a3 = fmaf(d3, d3, a3);
  }
  const float acc = (a0 + a1) + (a2 + a3);

  // Wave32 reduction via f32 WMMA (D = A x ones + 0).
  v2f a;
  a.x = acc;
  a.y = 0.0f;
  v2f b;
  b.x = 1.0f;
  b.y = 1.0f;
  v8f cm = {};
  cm = __builtin_amdgcn_wmma_f32_16x16x4_f32(
      /*neg_a=*/false, a, /*neg_b=*/false, b,
      /*c_mod=*/(short)0, cm, /*reuse_a=*/false, /*reuse_b=*/false);
  float s = ((cm[0] + cm[1]) + (cm[2] + cm[3])) +
            ((cm[4] + cm[5]) + (cm[6] + cm[7]));
  // Swap the two half-waves (lanes 0-15 <-> 16-31) and combine.
  s += __int_as_float(
      __builtin_amdgcn_ds_swizzle(__float_as_int(s), SWIZ_SWAPX16));

  __shared__ float lds[TPB / 32];
  const int lane = threadIdx.x & 31;
  const int wave = threadIdx.x >> 5;
  if (lane == 0) lds[wave] = s;
  __syncthreads();
  if (threadIdx.x == 0) {
    float t = 0.0f;
    for (int w = 0; w < TPB / 32; ++w) t += lds[w];
    partial[blockIdx.x] = t;  // fixed-order -> deterministic
  }
}

// ---------------------------------------------------------------------------
// Kernel 3: deterministic final tree reduction of NBLK partials in double,
// scaled by 1/(H*W*B), written to d_out[0] as float.
// ---------------------------------------------------------------------------
__global__ __launch_bounds__(1024) void kp_final_kernel(
    const float* __restrict__ partial, float* __restrict__ out) {
  __shared__ double lds[1024];
  double t = 0.0;
  for (int i = threadIdx.x; i < NBLK; i += 1024) t += (double)partial[i];
  lds[threadIdx.x] = t;
  __syncthreads();
  for (int s = 512; s > 0; s >>= 1) {
    if ((int)threadIdx.x < s) lds[threadIdx.x] += lds[threadIdx.x + s];
    __syncthreads();
  }
  if (threadIdx.x == 0) {
    out[0] = (float)(lds[0] * (1.0 / ((double)HH * (double)WW * (double)BB)));
  }
}

extern "C" void kernel_launch(void* const* d_in, const int* in_sizes, int n_in,
                              void* d_out, int out_size, void* d_ws,
                              size_t ws_size, hipStream_t stream) {
  const float* output = (const float*)d_in[0];  // (B,K,H,W) f32
  const float* target = (const float*)d_in[1];  // (B,K,2)   f32

  float* FX = (float*)d_ws;            // BK*HH floats
  float* FY = FX + BK * HH;            // BK*WW floats
  float* partial = FY + BK * WW;       // NBLK floats  (~545 KB total)

  kp_factors_kernel<<<BK, TPB, 0, stream>>>(target, FX, FY);
  kp_reduce_kernel<<<NBLK, TPB, 0, stream>>>(output, FX, FY, partial);
  kp_final_kernel<<<1, 1024, 0, stream>>>(partial, (float*)d_out);
}